// ChamferLoss_76029511074480
// MI455X (gfx1250) — compile-verified
//
#include <hip/hip_runtime.h>
#include <hip/hip_bf16.h>
#include <math.h>

// ---------------------------------------------------------------------------
// Chamfer loss for MI455X (gfx1250, wave32) -- single fused pass.
//
// dist(n,m) = |p_n|^2 + |g_m|^2 - 2 p_n.g_m computed ONCE in 16x16 tiles via
// V_WMMA_F32_16X16X4_F32:
//   A (16x4) : row n = (-2x, -2y, -2z, 1)        -- per-wave constant
//   B (4x16) : col m = ( x,   y,   z,  g2_m)     -- staged in LDS
//   C (16x16): p2_n broadcast per row            -- loop-invariant registers
//   D        = exact squared distance tile.
// Row mins (pred->gt) accumulate in registers via v_min3_num_f32.
// Col mins (gt->pred) reduce 8 components/lane then LDS atomic-min into a
// per-WG colmin[N]; merged to global minG with float atomicMin at the end.
// ---------------------------------------------------------------------------

typedef float v2f __attribute__((ext_vector_type(2)));
typedef float v8f __attribute__((ext_vector_type(8)));

#define ROWS_PER_WAVE  16
#define WAVES_PER_WG   8
#define ROWS_PER_WG    (ROWS_PER_WAVE * WAVES_PER_WG)   // 128
#define LDS_TILE_M     256                               // gt points per stage
#define NN_BLOCK       (WAVES_PER_WG * 32)               // 256 threads
#define MAX_N          8192
#define FLT_BIG        3.4e38f

__global__ __launch_bounds__(NN_BLOCK)
void chamfer_fused_kernel(const float* __restrict__ P,     // pred (B,N,3)
                          const float* __restrict__ G,     // gt   (B,N,3)
                          float* __restrict__ minP,        // (B,N) row mins
                          float* __restrict__ minG,        // (B,N) col mins (pre-init +INF)
                          int N, int blocksPerBatch) {
    __shared__ float2 sg[2 * LDS_TILE_M];   // staged gt: (x,y) then (z,g2)
    __shared__ float  colmin[MAX_N];        // per-WG col partial mins

    const int bx      = blockIdx.x;
    const int batch   = bx / blocksPerBatch;
    const int tile    = bx % blocksPerBatch;
    const int tid     = threadIdx.x;
    const int lane    = tid & 31;
    const int wave    = tid >> 5;
    const int rowBase = tile * ROWS_PER_WG + wave * ROWS_PER_WAVE;
    const bool hiHalf = lane >= 16;

    // Init per-WG column mins.
    for (int i = tid; i < N; i += NN_BLOCK) colmin[i] = FLT_BIG;

    // ---- A operand (16x4 f32): lanes 0-15 hold K0,K1; lanes 16-31 K2,K3 ---
    const int myRow = rowBase + (lane & 15);
    const float* pa = P + ((size_t)batch * N + myRow) * 3;
    const float ax = pa[0], ay = pa[1], az = pa[2];
    const float p2own = ax * ax + ay * ay + az * az;
    v2f aop;
    aop[0] = hiHalf ? (-2.0f * az) : (-2.0f * ax);
    aop[1] = hiHalf ? 1.0f         : (-2.0f * ay);

    // ---- C operand: p2 of row (v + hi*8), identical across the 16 lanes ---
    v8f cvec;
#pragma unroll
    for (int v = 0; v < 8; ++v)
        cvec[v] = __shfl(p2own, v + (hiHalf ? 8 : 0), 32);

    v8f rowmin;
#pragma unroll
    for (int k = 0; k < 8; ++k) rowmin[k] = FLT_BIG;

    const float* Gb = G + (size_t)batch * N * 3;
    const float2* myCol = &sg[(hiHalf ? LDS_TILE_M : 0) + (lane & 15)];

    __syncthreads();   // colmin init visible

    for (int mBase = 0; mBase < N; mBase += LDS_TILE_M) {
        __syncthreads();   // previous tile fully consumed
        {
            const int m = mBase + tid;
            const float* pg = Gb + (size_t)m * 3;
            const float gx = pg[0], gy = pg[1], gz = pg[2];
            float2 lo, hi;
            lo.x = gx; lo.y = gy;
            hi.x = gz; hi.y = gx * gx + gy * gy + gz * gz;   // (z, g2)
            sg[tid] = lo;
            sg[LDS_TILE_M + tid] = hi;
        }
        __syncthreads();

#pragma unroll 2
        for (int pair = 0; pair < LDS_TILE_M / 32; ++pair) {
            const float2 q0 = myCol[(2 * pair + 0) * 16];   // ds_load_2addr_b64
            const float2 q1 = myCol[(2 * pair + 1) * 16];
            v2f b0; b0[0] = q0.x; b0[1] = q0.y;
            v2f b1; b1[0] = q1.x; b1[1] = q1.y;
            // Exact distance tiles: D = A*B + p2_row
            v8f d0 = __builtin_amdgcn_wmma_f32_16x16x4_f32(
                false, aop, false, b0, (short)0, cvec, false, false);
            v8f d1 = __builtin_amdgcn_wmma_f32_16x16x4_f32(
                false, aop, false, b1, (short)0, cvec, false, false);

            // Row mins: rowmin = min(d0, d1, rowmin), no fcanonicalize.
#pragma unroll
            for (int k = 0; k < 8; ++k) {
                float r = rowmin[k];
                asm("v_min3_num_f32 %0, %1, %2, %0"
                    : "+v"(r) : "v"(d0[k]), "v"(d1[k]));
                rowmin[k] = r;
            }

            // Col mins: reduce 8 rows held in this lane, then LDS atomic-min.
            float c0, c1;
            asm("v_min3_num_f32 %0, %1, %2, %3" : "=v"(c0)
                : "v"(d0[0]), "v"(d0[1]), "v"(d0[2]));
            asm("v_min3_num_f32 %0, %0, %1, %2" : "+v"(c0)
                : "v"(d0[3]), "v"(d0[4]));
            asm("v_min3_num_f32 %0, %0, %1, %2" : "+v"(c0)
                : "v"(d0[5]), "v"(d0[6]));
            asm("v_min_num_f32 %0, %0, %1"      : "+v"(c0) : "v"(d0[7]));
            asm("v_min3_num_f32 %0, %1, %2, %3" : "=v"(c1)
                : "v"(d1[0]), "v"(d1[1]), "v"(d1[2]));
            asm("v_min3_num_f32 %0, %0, %1, %2" : "+v"(c1)
                : "v"(d1[3]), "v"(d1[4]));
            asm("v_min3_num_f32 %0, %0, %1, %2" : "+v"(c1)
                : "v"(d1[5]), "v"(d1[6]));
            asm("v_min_num_f32 %0, %0, %1"      : "+v"(c1) : "v"(d1[7]));

            const int col = lane & 15;
            atomicMin(&colmin[mBase + pair * 32 + col],      c0);
            atomicMin(&colmin[mBase + pair * 32 + 16 + col], c1);
        }
    }

    // ---- Row mins: reduce across 16 columns (xor within 16-lane groups) ---
#pragma unroll
    for (int off = 8; off >= 1; off >>= 1) {
#pragma unroll
        for (int k = 0; k < 8; ++k)
            rowmin[k] = __builtin_fminf(rowmin[k], __shfl_xor(rowmin[k], off, 32));
    }

    // lanes 0..7 -> rows 0..7 (comp=lane); lanes 24..31 -> rows 8..15.
    int comp = -1, row = 0;
    if (lane < 8)        { comp = lane;      row = lane; }
    else if (lane >= 24) { comp = lane - 24; row = lane - 16; }
    if (comp >= 0) {
        float mv = rowmin[0];
#pragma unroll
        for (int k = 1; k < 8; ++k) mv = (comp == k) ? rowmin[k] : mv;
        minP[(size_t)batch * N + rowBase + row] = mv;
    }

    // ---- Merge per-WG col mins into global minG (order-independent) ------
    __syncthreads();
    for (int i = tid; i < N; i += NN_BLOCK)
        atomicMin(&minG[(size_t)batch * N + i], colmin[i]);
}

__global__ __launch_bounds__(256)
void fill_inf_kernel(float* __restrict__ p, int n) {
    const int i = blockIdx.x * 256 + threadIdx.x;
    if (i < n) p[i] = FLT_BIG;
}

// loss = mean_{b,n} (minP + minG) * mean_d(weight)
__global__ __launch_bounds__(256)
void chamfer_reduce_kernel(const float* __restrict__ minP,
                           const float* __restrict__ minG,
                           const float* __restrict__ w,
                           float* __restrict__ out, int total) {
    float s = 0.0f;
    for (int i = threadIdx.x; i < total; i += 256) {
        const float wm = (w[3 * i] + w[3 * i + 1] + w[3 * i + 2]) * (1.0f / 3.0f);
        s += (minP[i] + minG[i]) * wm;
    }
#pragma unroll
    for (int off = 16; off >= 1; off >>= 1) s += __shfl_xor(s, off, 32);

    __shared__ float ls[8];
    const int lane = threadIdx.x & 31, wv = threadIdx.x >> 5;
    if (lane == 0) ls[wv] = s;
    __syncthreads();
    if (wv == 0) {
        float t = (lane < 8) ? ls[lane] : 0.0f;
#pragma unroll
        for (int off = 4; off >= 1; off >>= 1) t += __shfl_xor(t, off, 32);
        if (lane == 0) out[0] = t / (float)total;
    }
}

extern "C" void kernel_launch(void* const* d_in, const int* in_sizes, int n_in,
                              void* d_out, int out_size, void* d_ws, size_t ws_size,
                              hipStream_t stream) {
    const float* pred   = (const float*)d_in[0];  // inputs  (B,N,3)
    const float* gt     = (const float*)d_in[1];  // targets (B,N,3)
    const float* weight = (const float*)d_in[2];  // weight  (B,N,3)
    float* out = (float*)d_out;

    const int B = 4;
    const int total = in_sizes[0] / 3;   // B*N
    const int N = total / B;             // 8192

    float* minP = (float*)d_ws;          // (B,N) fully written by fused kernel
    float* minG = minP + total;          // (B,N) atomic-min target

    fill_inf_kernel<<<(total + 255) / 256, 256, 0, stream>>>(minG, total);

    const int blocksPerBatch = N / ROWS_PER_WG;          // 64
    dim3 grid(B * blocksPerBatch);                       // 256 workgroups
    dim3 block(NN_BLOCK);                                // 256 threads = 8 waves
    chamfer_fused_kernel<<<grid, block, 0, stream>>>(pred, gt, minP, minG,
                                                     N, blocksPerBatch);

    chamfer_reduce_kernel<<<1, 256, 0, stream>>>(minP, minG, weight, out, total);
}